// BatchDecoder_25340307047174
// MI455X (gfx1250) — compile-verified
//
#include <hip/hip_runtime.h>
#include <hip/hip_bf16.h>

typedef __attribute__((ext_vector_type(2))) float v2f;
typedef __attribute__((ext_vector_type(8))) float v8f;

#define NUM_EMB  64
#define X_SIZE   128
#define H_SIZE   256
#define OUT_SIZE 128
#define EMBED_DIM (H_SIZE*X_SIZE + H_SIZE + OUT_SIZE*H_SIZE + OUT_SIZE) // 65920
#define B1_OFF (H_SIZE*X_SIZE)            // 32768
#define W2_OFF (B1_OFF + H_SIZE)          // 33024
#define B2_OFF (W2_OFF + OUT_SIZE*H_SIZE) // 65792
#define TILE_N 32
#define XPAD 132   // 128 + pad -> avoids 64-bank conflicts on strided column reads
#define HPAD 260   // 256 + pad

// ws layout (ints): [0,64) counts, [64,128) rep, [128,193) starts, [256, 256+n) order
__global__ void BatchDecoder_group_kernel(const int* __restrict__ emb_idx, int n,
                                          int* __restrict__ ws) {
    int e = threadIdx.x;                 // 0..63
    int* counts = ws;
    int* rep    = ws + 64;
    int* starts = ws + 128;
    int* order  = ws + 256;

    int c = 0, r = 0, found = 0;
    for (int i = 0; i < n; ++i) {
        if (emb_idx[i] == e) { if (!found) { r = i; found = 1; } ++c; }
    }
    counts[e] = c;
    rep[e]    = found ? r : 0;
    __syncthreads();
    if (e == 0) {
        int s = 0;
        for (int g = 0; g < NUM_EMB; ++g) { starts[g] = s; s += counts[g]; }
        starts[NUM_EMB] = s;
    }
    __syncthreads();
    int off = starts[e];
    for (int i = 0; i < n; ++i) {
        if (emb_idx[i] == e) order[off++] = i;
    }
}

// One block = one (embedding group, 32-sample tile).
__global__ __launch_bounds__(256) void BatchDecoder_mlp_kernel(
        const float* __restrict__ qf,   // quant_fn  [n, EMBED_DIM]
        const float* __restrict__ x,    // [n, X_SIZE]
        float* __restrict__ out,        // [n, OUT_SIZE]
        const int* __restrict__ ws) {
    const int* counts = ws;
    const int* rep    = ws + 64;
    const int* starts = ws + 128;
    const int* order  = ws + 256;

    const int e    = blockIdx.y;
    const int tile = blockIdx.x;
    const int cnt  = counts[e];
    if (tile * TILE_N >= cnt) return;                 // uniform exit
    const int base   = starts[e] + tile * TILE_N;
    const int nvalid = min(TILE_N, cnt - tile * TILE_N);
    const float* __restrict__ W = qf + (size_t)rep[e] * EMBED_DIM;

    __shared__ float xs[TILE_N][XPAD];
    __shared__ float hs[TILE_N][HPAD];
    __shared__ int   sid[TILE_N];

    const int t = threadIdx.x;
    if (t < TILE_N) {
        int j = (t < nvalid) ? t : 0;                 // pad with first sample (recomputed, not stored)
        sid[t] = order[base + j];
    }
    __syncthreads();

    // Stage X tile: thread t -> sample t>>3, 16 floats at chunk (t&7)*16
    {
        const int j = t >> 3;
        const int c = (t & 7) * 16;
        const float* __restrict__ xr = x + (size_t)sid[j] * X_SIZE + c;
        #pragma unroll
        for (int i = 0; i < 16; ++i) xs[j][c + i] = xr[i];
    }
    __syncthreads();

    const int wave = t >> 5;
    const int lane = t & 31;
    const int col  = lane & 15;          // N-column (feature) this lane owns in B/D
    const int ksel = (lane >> 4) * 2;    // K sub-offset per ISA A/B fragment layout
    const int mhi  = 8 * (lane >> 4);    // D-row offset for this lane half

    // ---------------- Layer 1: H = relu(X @ W1^T + b1) ----------------
    // 2 sample-subtiles x 16 h-tiles = 32 (16x16) tiles; wave w -> subtile w>>2, h-tiles (w&3)*4 ..+3
    {
        const int s  = wave >> 2;
        const int h4 = (wave & 3) * 4;
        v8f acc[4] = {};
        const float* __restrict__ arow = &xs[s * 16 + col][0];
        for (int k = 0; k < X_SIZE; k += 4) {
            v2f a; a.x = arow[k + ksel]; a.y = arow[k + ksel + 1];
            #pragma unroll
            for (int i = 0; i < 4; ++i) {
                const int h = (h4 + i) * 16 + col;
                const float* __restrict__ wr = W + (size_t)h * X_SIZE + k + ksel; // W1 row-major
                v2f b; b.x = wr[0]; b.y = wr[1];
                acc[i] = __builtin_amdgcn_wmma_f32_16x16x4_f32(
                             false, a, false, b, (short)0, acc[i], false, false);
            }
        }
        #pragma unroll
        for (int i = 0; i < 4; ++i) {
            const int h = (h4 + i) * 16 + col;
            const float bias = W[B1_OFF + h];
            #pragma unroll
            for (int r = 0; r < 8; ++r) {
                hs[s * 16 + r + mhi][h] = fmaxf(acc[i][r] + bias, 0.0f);
            }
        }
    }
    __syncthreads();

    // ---------------- Layer 2: out = H @ W2^T + b2 ----------------
    // 2 sample-subtiles x 8 o-tiles = 16 tiles; wave w -> subtile w>>2, o-tiles (w&3)*2 ..+1
    {
        const int s  = wave >> 2;
        const int o2 = (wave & 3) * 2;
        v8f acc[2] = {};
        const float* __restrict__ arow = &hs[s * 16 + col][0];
        for (int k = 0; k < H_SIZE; k += 4) {
            v2f a; a.x = arow[k + ksel]; a.y = arow[k + ksel + 1];
            #pragma unroll
            for (int i = 0; i < 2; ++i) {
                const int o = (o2 + i) * 16 + col;
                const float* __restrict__ wr = W + W2_OFF + (size_t)o * H_SIZE + k + ksel; // W2 row-major
                v2f b; b.x = wr[0]; b.y = wr[1];
                acc[i] = __builtin_amdgcn_wmma_f32_16x16x4_f32(
                             false, a, false, b, (short)0, acc[i], false, false);
            }
        }
        #pragma unroll
        for (int i = 0; i < 2; ++i) {
            const int o = (o2 + i) * 16 + col;
            const float bias = W[B2_OFF + o];
            #pragma unroll
            for (int r = 0; r < 8; ++r) {
                const int m = s * 16 + r + mhi;     // tile-local sample
                if (m < nvalid) {
                    out[(size_t)sid[m] * OUT_SIZE + o] = acc[i][r] + bias;
                }
            }
        }
    }
}

extern "C" void kernel_launch(void* const* d_in, const int* in_sizes, int n_in,
                              void* d_out, int out_size, void* d_ws, size_t ws_size,
                              hipStream_t stream) {
    const float* quant_fn = (const float*)d_in[0];
    const float* x        = (const float*)d_in[1];
    const int*   emb_idx  = (const int*)d_in[2];
    float*       out      = (float*)d_out;
    int*         ws       = (int*)d_ws;
    const int n = in_sizes[2];

    BatchDecoder_group_kernel<<<1, 64, 0, stream>>>(emb_idx, n, ws);

    const int max_tiles = (n + TILE_N - 1) / TILE_N;   // covers worst-case skew (all samples in one group)
    dim3 grid(max_tiles, NUM_EMB);
    BatchDecoder_mlp_kernel<<<grid, 256, 0, stream>>>(quant_fn, x, out, ws);
}